// MoERouter_5308579577969
// MI455X (gfx1250) — compile-verified
//
#include <hip/hip_runtime.h>

// MoE router for MI455X (gfx1250, wave32, WMMA + async global->LDS).
// final[b,:] = sum_e w[b,e] * (x[b,:] @ W_e + bias_e); w dense with zeros for
// unselected experts -> one fp32 WMMA accumulation chain per 16x16 tile.

typedef __attribute__((ext_vector_type(2))) float v2f;
typedef __attribute__((ext_vector_type(8))) float v8f;

#define BATCH 16384
#define CTX   2048
#define HOR   64
#define NEXP  6

// ---------------------------------------------------------------------------
// Kernel 1: gating. One wave32 per sample: 6 dot products of length 2048,
// butterfly reduction, softmax over 6, top-2, dense weight row (B x 6).
// ---------------------------------------------------------------------------
__global__ __launch_bounds__(256) void moe_gating_kernel(
    const float* __restrict__ x, const float* __restrict__ gw,
    const float* __restrict__ gb, float* __restrict__ wdense)
{
    const int lane = threadIdx.x & 31;
    const int wave = threadIdx.x >> 5;
    const int b = blockIdx.x * 8 + wave;
    const float* xr = x + (size_t)b * CTX;

    float acc[NEXP];
#pragma unroll
    for (int e = 0; e < NEXP; ++e) acc[e] = 0.0f;

    for (int i = lane; i < CTX; i += 32) {
        float xv = xr[i];
#pragma unroll
        for (int e = 0; e < NEXP; ++e)
            acc[e] = fmaf(xv, gw[e * CTX + i], acc[e]);
    }
#pragma unroll
    for (int e = 0; e < NEXP; ++e) {
        float v = acc[e];
#pragma unroll
        for (int off = 16; off > 0; off >>= 1) v += __shfl_xor(v, off, 32);
        acc[e] = v;
    }
    float logit[NEXP], p[NEXP];
    float mx = -1e30f;
#pragma unroll
    for (int e = 0; e < NEXP; ++e) { logit[e] = acc[e] + gb[e]; mx = fmaxf(mx, logit[e]); }
    float s = 0.0f;
#pragma unroll
    for (int e = 0; e < NEXP; ++e) { p[e] = expf(logit[e] - mx); s += p[e]; }
    float inv = 1.0f / s;
#pragma unroll
    for (int e = 0; e < NEXP; ++e) p[e] *= inv;

    int i1 = 0; float v1 = p[0];
#pragma unroll
    for (int e = 1; e < NEXP; ++e) if (p[e] > v1) { v1 = p[e]; i1 = e; }
    int i2 = -1; float v2 = -1.0f;
#pragma unroll
    for (int e = 0; e < NEXP; ++e) if (e != i1 && p[e] > v2) { v2 = p[e]; i2 = e; }

    float s2 = v1 + v2;
    float w1 = v1 / (s2 + 1e-8f);
    float w2 = v2 / (s2 + 1e-8f);
    if (lane < NEXP) {
        float wvv = (lane == i1) ? w1 : ((lane == i2) ? w2 : 0.0f);
        wdense[b * NEXP + lane] = wvv;
    }
}

// ---------------------------------------------------------------------------
// Kernel 2: experts + combine via V_WMMA_F32_16X16X4_F32, with x staged
// through LDS by GLOBAL_LOAD_ASYNC_TO_LDS_B128 (ASYNCcnt), double buffered.
// Block = 128 threads (4 waves) per 16-row tile; wave w owns columns
// [16w, 16w+16). Gating weight folded into the A fragment; bias folded in
// as one rank-1 WMMA per expert.
// ---------------------------------------------------------------------------
#define KCHUNK 256
#define NCHUNK (CTX / KCHUNK)            // 8
#define LDS_STRIDE (KCHUNK + 4)          // 1040 B row stride => rows 4 banks apart
#define XFERS_PER_THREAD ((16 * KCHUNK) / (128 * 4))   // 8 x b128 per thread

__device__ __forceinline__ void async_stage_chunk(
    const float* __restrict__ xrow, float* __restrict__ ldsbuf,
    int tid, int kc)
{
#pragma unroll
    for (int it = 0; it < XFERS_PER_THREAD; ++it) {
        int v = tid + 128 * it;
        int f = v * 4;
        int r = f >> 8;                   // / KCHUNK
        int col = f & (KCHUNK - 1);
        const float* gaddr = xrow + (size_t)r * CTX + kc + col;
        unsigned lds_off = (unsigned)(uintptr_t)(&ldsbuf[r * LDS_STRIDE + col]);
        asm volatile("global_load_async_to_lds_b128 %0, %1, off"
                     :: "v"(lds_off), "v"(gaddr) : "memory");
    }
}

__global__ __launch_bounds__(128) void moe_expert_kernel(
    const float* __restrict__ x, const float* __restrict__ ew,
    const float* __restrict__ eb, const float* __restrict__ wdense,
    float* __restrict__ out)
{
    __shared__ float lds[2][16 * LDS_STRIDE];   // 33,280 B total

    const int tid  = threadIdx.x;
    const int lane = tid & 31;
    const int wv   = tid >> 5;       // N-tile 0..3
    const int half = lane >> 4;      // 0: K={0,1}, 1: K={2,3}
    const int l15  = lane & 15;
    const int rowbase = blockIdx.x * 16;
    const int nbase   = wv * 16;

    // gating weights for this lane's A-row (row = l15)
    float wl[NEXP];
#pragma unroll
    for (int e = 0; e < NEXP; ++e)
        wl[e] = wdense[(rowbase + l15) * NEXP + e];

    v8f c = {};   // 16x16 f32 accumulator (8 VGPRs)

    const float* xrow = x + (size_t)rowbase * CTX;

    // prologue: fill buffer 0 with chunk 0
    async_stage_chunk(xrow, &lds[0][0], tid, 0);

    for (int ci = 0; ci < NCHUNK; ++ci) {
        const int buf = ci & 1;
        const int kc = ci * KCHUNK;
        // my async loads into lds[buf] are complete:
        asm volatile("s_wait_asynccnt 0x0" ::: "memory");
        // all waves' loads done AND everyone finished reading lds[buf^1]:
        __syncthreads();
        // stream next chunk into the other buffer while we compute
        if (ci + 1 < NCHUNK)
            async_stage_chunk(xrow, &lds[(ci + 1) & 1][0], tid, kc + KCHUNK);

#pragma unroll 1
        for (int e = 0; e < NEXP; ++e) {
            const float we = wl[e];
            const float* bp = ew + ((size_t)e * CTX + (size_t)kc + 2 * half) * HOR + nbase + l15;
            const float* ap = &lds[buf][l15 * LDS_STRIDE + 2 * half];
#pragma unroll 4
            for (int k0 = 0; k0 < KCHUNK; k0 += 4) {
                v2f a = *(const v2f*)(ap + k0);       // ds_load_b64, conflict-free
                a.x *= we; a.y *= we;                 // fold gating weight into A
                v2f bb;
                bb.x = bp[(size_t)k0 * HOR];          // W_e[k,   n]
                bb.y = bp[(size_t)k0 * HOR + HOR];    // W_e[k+1, n]
                c = __builtin_amdgcn_wmma_f32_16x16x4_f32(
                        false, a, false, bb, (short)0, c, false, false);
            }
        }
    }

    // bias: rank-1 update per expert, D += w[m,e] * bias_e[n]
#pragma unroll 1
    for (int e = 0; e < NEXP; ++e) {
        v2f a, bb;
        a.x = (half == 0) ? wl[e] : 0.0f;
        a.y = 0.0f;
        bb.x = (half == 0) ? eb[e * HOR + nbase + l15] : 0.0f;
        bb.y = 0.0f;
        c = __builtin_amdgcn_wmma_f32_16x16x4_f32(
                false, a, false, bb, (short)0, c, false, false);
    }

    // store: VGPR j -> M = j + 8*half, N = l15 (ISA 7.12.2 C/D layout)
#pragma unroll
    for (int j = 0; j < 8; ++j) {
        int m = rowbase + j + 8 * half;
        out[(size_t)m * HOR + nbase + l15] = c[j];
    }
}

// ---------------------------------------------------------------------------
extern "C" void kernel_launch(void* const* d_in, const int* in_sizes, int n_in,
                              void* d_out, int out_size, void* d_ws, size_t ws_size,
                              hipStream_t stream) {
    const float* x  = (const float*)d_in[0];   // (B, CTX)
    const float* gw = (const float*)d_in[1];   // (E, CTX)
    const float* gb = (const float*)d_in[2];   // (E)
    const float* ew = (const float*)d_in[3];   // (E, CTX, H)
    const float* eb = (const float*)d_in[4];   // (E, H)
    float* wdense = (float*)d_ws;              // (B, E) dense gating weights
    float* out = (float*)d_out;                // (B, H)

    moe_gating_kernel<<<BATCH / 8, 256, 0, stream>>>(x, gw, gb, wdense);
    moe_expert_kernel<<<BATCH / 16, 128, 0, stream>>>(x, ew, eb, wdense, out);
}